// LinkPredictor_4123168604526
// MI455X (gfx1250) — compile-verified
//
#include <hip/hip_runtime.h>
#include <hip/hip_bf16.h>

typedef __attribute__((ext_vector_type(2))) float v2f;
typedef __attribute__((ext_vector_type(8))) float v8f;

#define DIM 128
#define NCOLS 256
#define EDGE_BLOCKS 1024
#define EDGE_THREADS 256

// ---------------------------------------------------------------------------
// Kernel 0: build Wmat[k][i] (row-major 128x256) so the GEMM reads B linearly.
//   i < 128 :  Wmat[k][i] = W1[i][k]        (A half: x @ W1[:, :128].T)
//   i >= 128:  Wmat[k][i] = W1[i-128][k+128] (B half: x @ W1[:, 128:].T)
// W1 is row-major (128, 256).
// ---------------------------------------------------------------------------
__global__ void prep_wmat_kernel(const float* __restrict__ W1, float* __restrict__ Wmat) {
    int idx = blockIdx.x * blockDim.x + threadIdx.x;
    if (idx >= DIM * NCOLS) return;
    int k = idx >> 8;     // 0..127  (input dim)
    int i = idx & 255;    // 0..255  (output col)
    float v = (i < DIM) ? W1[i * NCOLS + k] : W1[(i - DIM) * NCOLS + (k + DIM)];
    Wmat[idx] = v;
}

// ---------------------------------------------------------------------------
// Kernel 1: Y = x @ Wmat  (M=n_nodes, K=128, N=256) via V_WMMA_F32_16X16X4_F32.
// Block = 128 threads = 4 waves. Block computes a 16-row strip; wave w owns
// columns [w*64, w*64+64) as four 16x16 accumulators (A fragment reused 4x).
// b1 is folded into columns 0..127 at store time.
// ---------------------------------------------------------------------------
__global__ void gemm_wmma_kernel(const float* __restrict__ x,
                                 const float* __restrict__ Wmat,
                                 const float* __restrict__ b1,
                                 float* __restrict__ Y,
                                 int n_nodes) {
    const int lane  = threadIdx.x & 31;
    const int wave  = threadIdx.x >> 5;
    const int m0    = blockIdx.x * 16;
    const int mrow  = lane & 15;           // M within tile (A fragment & store)
    const int khalf = (lane >> 4) << 1;    // 0 or 2: which K-pair this lane holds
    const int ncol  = lane & 15;           // N within 16-wide tile (B fragment)
    const int nbase = wave * 64;

    v8f acc0 = {}, acc1 = {}, acc2 = {}, acc3 = {};

    int node = m0 + mrow;
    if (node >= n_nodes) node = n_nodes - 1;   // clamp reads (n_nodes % 16 == 0 normally)
    const float* xrow = x + (size_t)node * DIM;

    #pragma unroll 4
    for (int k0 = 0; k0 < DIM; k0 += 4) {
        v2f a;
        a.x = xrow[k0 + khalf];
        a.y = xrow[k0 + khalf + 1];
        const float* w0 = Wmat + (size_t)(k0 + khalf)     * NCOLS + nbase + ncol;
        const float* w1 = Wmat + (size_t)(k0 + khalf + 1) * NCOLS + nbase + ncol;
        v2f b;
        b.x = w0[0];  b.y = w1[0];
        acc0 = __builtin_amdgcn_wmma_f32_16x16x4_f32(false, a, false, b, (short)0, acc0, false, false);
        b.x = w0[16]; b.y = w1[16];
        acc1 = __builtin_amdgcn_wmma_f32_16x16x4_f32(false, a, false, b, (short)0, acc1, false, false);
        b.x = w0[32]; b.y = w1[32];
        acc2 = __builtin_amdgcn_wmma_f32_16x16x4_f32(false, a, false, b, (short)0, acc2, false, false);
        b.x = w0[48]; b.y = w1[48];
        acc3 = __builtin_amdgcn_wmma_f32_16x16x4_f32(false, a, false, b, (short)0, acc3, false, false);
    }

    // C/D layout: VGPR r, lane l -> M = r + 8*(l>=16), N = l%16.
    const int mbase = m0 + ((lane >> 4) << 3);
    #pragma unroll
    for (int t = 0; t < 4; ++t) {
        const v8f acc = (t == 0) ? acc0 : (t == 1) ? acc1 : (t == 2) ? acc2 : acc3;
        const int col = nbase + t * 16 + ncol;
        const float bias = (col < DIM) ? b1[col] : 0.0f;   // fold b1 into A-half
        #pragma unroll
        for (int r = 0; r < 8; ++r) {
            int m = mbase + r;
            if (m < n_nodes) Y[(size_t)m * NCOLS + col] = acc[r] + bias;
        }
    }
}

// ---------------------------------------------------------------------------
// Kernel 2: edge phase. One wave32 per edge (grid-stride):
//   z = dot(relu(Y[u, 0:128] + Y[v, 128:256]), W2) + b2
//   loss_e = log1p(exp(-|z|)) + y*max(-z,0) + (1-y)*max(z,0)
// Deterministic reduction: wave shfl_xor -> LDS -> per-block partial.
// ---------------------------------------------------------------------------
__global__ void edge_loss_kernel(const float* __restrict__ Y,
                                 const float* __restrict__ W2,
                                 const float* __restrict__ b2,
                                 const float* __restrict__ labels,
                                 const int* __restrict__ pairs,
                                 float* __restrict__ partials,
                                 int E) {
    const int lane   = threadIdx.x & 31;
    const int widx   = threadIdx.x >> 5;
    const int gwave  = (blockIdx.x * blockDim.x + threadIdx.x) >> 5;
    const int nwaves = (gridDim.x * blockDim.x) >> 5;

    const float4 w2v = ((const float4*)W2)[lane];   // 32 lanes * 4 = 128 weights
    const float  b2v = b2[0];

    float acc = 0.0f;
    for (int e = gwave; e < E; e += nwaves) {
        const int u = pairs[e];
        const int v = pairs[E + e];
        const float4 au = ((const float4*)(Y + (size_t)u * NCOLS))[lane];
        const float4 bv = ((const float4*)(Y + (size_t)v * NCOLS + DIM))[lane];
        const float h0 = fmaxf(au.x + bv.x, 0.0f);
        const float h1 = fmaxf(au.y + bv.y, 0.0f);
        const float h2 = fmaxf(au.z + bv.z, 0.0f);
        const float h3 = fmaxf(au.w + bv.w, 0.0f);
        float s = h0 * w2v.x + h1 * w2v.y + h2 * w2v.z + h3 * w2v.w;
        s += __shfl_xor(s, 16, 32);
        s += __shfl_xor(s,  8, 32);
        s += __shfl_xor(s,  4, 32);
        s += __shfl_xor(s,  2, 32);
        s += __shfl_xor(s,  1, 32);
        const float z = s + b2v;
        const float y = labels[e];
        const float loss = __logf(1.0f + __expf(-fabsf(z)))
                         + y * fmaxf(-z, 0.0f) + (1.0f - y) * fmaxf(z, 0.0f);
        acc += loss;   // every lane holds the same value; only lane 0's is used
    }

    __shared__ float lds[EDGE_THREADS / 32];
    if (lane == 0) lds[widx] = acc;
    __syncthreads();
    if (threadIdx.x == 0) {
        float s = 0.0f;
        #pragma unroll
        for (int i = 0; i < EDGE_THREADS / 32; ++i) s += lds[i];
        partials[blockIdx.x] = s;
    }
}

// ---------------------------------------------------------------------------
// Kernel 3: final deterministic reduction of per-block partials, apply mean.
// ---------------------------------------------------------------------------
__global__ void final_reduce_kernel(const float* __restrict__ partials, int n,
                                    float* __restrict__ out, float invE) {
    __shared__ float lds[256];
    float s = 0.0f;
    for (int i = threadIdx.x; i < n; i += blockDim.x) s += partials[i];
    lds[threadIdx.x] = s;
    __syncthreads();
    for (int off = 128; off > 0; off >>= 1) {
        if ((int)threadIdx.x < off) lds[threadIdx.x] += lds[threadIdx.x + off];
        __syncthreads();
    }
    if (threadIdx.x == 0) out[0] = lds[0] * invE;
}

extern "C" void kernel_launch(void* const* d_in, const int* in_sizes, int n_in,
                              void* d_out, int out_size, void* d_ws, size_t ws_size,
                              hipStream_t stream) {
    const float* x      = (const float*)d_in[0];
    const float* W1     = (const float*)d_in[1];
    const float* b1     = (const float*)d_in[2];
    const float* W2     = (const float*)d_in[3];
    const float* b2     = (const float*)d_in[4];
    const float* labels = (const float*)d_in[5];
    const int*   pairs  = (const int*)d_in[6];

    const int n_nodes = in_sizes[0] / DIM;   // 50000
    const int E       = in_sizes[5];         // 600000

    // Workspace layout (bytes): Wmat [0, 128K) | Y [128K, 128K+nodes*256*4) | partials
    char* ws = (char*)d_ws;
    float* Wmat     = (float*)ws;                                        // 32768 f32
    float* Y        = (float*)(ws + (size_t)DIM * NCOLS * sizeof(float));// n_nodes*256 f32
    float* partials = (float*)(ws + (size_t)DIM * NCOLS * sizeof(float)
                                  + (size_t)n_nodes * NCOLS * sizeof(float));

    // 0) Build Wmat (linear-B layout for the WMMA GEMM)
    prep_wmat_kernel<<<(DIM * NCOLS + 255) / 256, 256, 0, stream>>>(W1, Wmat);

    // 1) Node-level GEMM via v_wmma_f32_16x16x4_f32: Y = x @ Wmat (+b1 folded)
    const int mblocks = (n_nodes + 15) / 16;
    gemm_wmma_kernel<<<mblocks, 128, 0, stream>>>(x, Wmat, b1, Y, n_nodes);

    // 2) Edge phase: L2-resident gathers of Y rows, per-block partial sums
    edge_loss_kernel<<<EDGE_BLOCKS, EDGE_THREADS, 0, stream>>>(
        Y, W2, b2, labels, pairs, partials, E);

    // 3) Deterministic final reduction + mean
    final_reduce_kernel<<<1, 256, 0, stream>>>(partials, EDGE_BLOCKS, (float*)d_out,
                                               1.0f / (float)E);
}